// TVMShieldedAttentionWithRPE_35682588295636
// MI455X (gfx1250) — compile-verified
//
#include <hip/hip_runtime.h>
#include <math.h>

// Problem constants (match reference)
#define B_  2
#define S_  2048
#define NK_ 32
#define H_  8
#define D_  64
#define DM_ 512
#define NEGC (-1000000000.0f)

typedef float v2f __attribute__((ext_vector_type(2)));
typedef float v8f __attribute__((ext_vector_type(8)));

// ---------------------------------------------------------------------------
// WMMA fp32 GEMM: C[M,N] = A[M,K] * Bm[K,N] (+ bias[N] + resid[M,N])
// Row-major everywhere. One wave (32 threads) computes a 32(M) x 64(N) tile.
// Uses V_WMMA_F32_16X16X4_F32 (exact fp32 MAC on the matrix pipe).
//
// f32 16x16x4 fragment layouts (CDNA5 ISA 7.12.2):
//   A (16x4):  lane l: m = l%16, kpair = l/16; {a.x,a.y} = A[m][k0+2*kpair+{0,1}]
//   B (4x16):  lane l: n = l%16, kpair = l/16; {b.x,b.y} = B[k0+2*kpair+{0,1}][n]
//   C/D (16x16): VGPR r: lanes 0-15 -> (M=r,   N=lane)
//                        lanes16-31 -> (M=r+8, N=lane-16)
// ---------------------------------------------------------------------------
__global__ void gemm_wmma_f32(const float* __restrict__ A,
                              const float* __restrict__ Bm,
                              const float* __restrict__ bias,
                              const float* __restrict__ resid,
                              float* __restrict__ C,
                              int M, int N, int K)
{
    const int lane  = threadIdx.x & 31;
    const int lm    = lane & 15;         // m (for A) / n (for B) within tile
    const int kpair = lane >> 4;         // 0 -> k{0,1}, 1 -> k{2,3}
    const int m0    = blockIdx.y * 32;
    const int n0    = blockIdx.x * 64;

    v8f acc[2][4];
    const v8f zero8 = {0.f,0.f,0.f,0.f,0.f,0.f,0.f,0.f};
#pragma unroll
    for (int i = 0; i < 2; ++i)
#pragma unroll
        for (int j = 0; j < 4; ++j)
            acc[i][j] = zero8;

    for (int k0 = 0; k0 < K; k0 += 4) {
        // --- A fragments for the two 16-row M tiles ---
        const float* ap = A + (size_t)(m0 + lm) * K + (k0 + 2 * kpair);
        v2f a0, a1;
        a0.x = ap[0];                a0.y = ap[1];
        a1.x = ap[(size_t)16 * K];   a1.y = ap[(size_t)16 * K + 1];

        // --- B fragments for the four 16-col N tiles ---
        const float* bp = Bm + (size_t)(k0 + 2 * kpair) * N + (n0 + lm);
        v2f bf[4];
#pragma unroll
        for (int j = 0; j < 4; ++j) {
            bf[j].x = bp[j * 16];
            bf[j].y = bp[(size_t)N + j * 16];
        }

#pragma unroll
        for (int j = 0; j < 4; ++j) {
            acc[0][j] = __builtin_amdgcn_wmma_f32_16x16x4_f32(
                false, a0, false, bf[j], (short)0, acc[0][j], false, false);
            acc[1][j] = __builtin_amdgcn_wmma_f32_16x16x4_f32(
                false, a1, false, bf[j], (short)0, acc[1][j], false, false);
        }
    }

    // --- Store (optionally fused bias + residual add) ---
    const int rowAdj = (lane < 16) ? 0 : 8;
    const int colL   = lm;
#pragma unroll
    for (int i = 0; i < 2; ++i) {
#pragma unroll
        for (int j = 0; j < 4; ++j) {
            const int colBase = n0 + j * 16 + colL;
#pragma unroll
            for (int r = 0; r < 8; ++r) {
                const int row = m0 + i * 16 + rowAdj + r;
                float v = acc[i][j][r];
                if (bias)  v += bias[colBase];
                if (resid) v += resid[(size_t)row * N + colBase];
                C[(size_t)row * N + colBase] = v;
            }
        }
    }
}

// ---------------------------------------------------------------------------
// Gathered attention with RPE. One block per (b,s): 8 waves = 8 heads,
// lane n = key index n (NK=32 == wave32). Softmax across the 32 lanes via
// shuffles; attn staged in LDS, then lanes flip to d-parallel for attn @ V.
// ---------------------------------------------------------------------------
__global__ void attn_gather_kernel(const float* __restrict__ Q,
                                   const float* __restrict__ Kt,
                                   const float* __restrict__ Vt,
                                   const float* __restrict__ rpe,
                                   const int*   __restrict__ qkm,
                                   float* __restrict__ ctx)
{
    __shared__ float q_s[DM_];      // all heads' q for this (b,s)
    __shared__ int   idx_s[NK_];
    __shared__ int   val_s[NK_];
    __shared__ float attn_s[H_ * NK_];

    const int bid = blockIdx.x;          // 0..B*S-1
    const int b   = bid >> 11;           // /S_
    const int s   = bid & (S_ - 1);
    const int t   = threadIdx.x;         // 0..255
    const int h   = t >> 5;              // wave id == head
    const int lane = t & 31;

    const size_t rowBase = (size_t)bid * DM_;
    q_s[t]       = Q[rowBase + t];
    q_s[t + 256] = Q[rowBase + t + 256];
    if (t < NK_) {
        int raw   = qkm[s * NK_ + t];
        val_s[t]  = (raw >= 0) ? 1 : 0;
        idx_s[t]  = (raw >= 0) ? raw : 0;
    }
    __syncthreads();

    // ---- Phase 1: scores + softmax (lane n = key n) ----
    const int n    = lane;
    const int valid = val_s[n];
    const size_t kRow = ((size_t)(b * S_ + idx_s[n]) * H_ + h) * D_;

    float dot = 0.f;
    const float4* kv = reinterpret_cast<const float4*>(Kt + kRow);
#pragma unroll
    for (int d4 = 0; d4 < D_ / 4; ++d4) {
        float4 kk = kv[d4];
        const int d = d4 * 4;
        dot += q_s[h * D_ + d + 0] * kk.x;
        dot += q_s[h * D_ + d + 1] * kk.y;
        dot += q_s[h * D_ + d + 2] * kk.z;
        dot += q_s[h * D_ + d + 3] * kk.w;
    }
    float sc = dot + rpe[((size_t)bid * NK_ + n) * H_ + h];
    sc = valid ? sc : NEGC;
    sc *= 0.125f;                         // 1/sqrt(64)

    float mx = sc;
#pragma unroll
    for (int off = 16; off > 0; off >>= 1)
        mx = fmaxf(mx, __shfl_xor(mx, off, 32));
    float e = expf(sc - mx) * (float)valid;
    float sum = e;
#pragma unroll
    for (int off = 16; off > 0; off >>= 1)
        sum += __shfl_xor(sum, off, 32);
    attn_s[h * NK_ + n] = (sum > 0.f) ? (e / sum) : 0.f;
    __syncthreads();

    // ---- Phase 2: ctx[h,d] = sum_n attn[n] * V[idx[n],h,d] (lane = d) ----
    float acc0 = 0.f, acc1 = 0.f;
#pragma unroll 4
    for (int j = 0; j < NK_; ++j) {
        float a = attn_s[h * NK_ + j];
        if (a != 0.f) {
            const size_t vRow = ((size_t)(b * S_ + idx_s[j]) * H_ + h) * D_;
            acc0 += a * Vt[vRow + lane];
            acc1 += a * Vt[vRow + lane + 32];
        }
    }
    const size_t oRow = ((size_t)bid * H_ + h) * D_;
    ctx[oRow + lane]      = acc0;
    ctx[oRow + lane + 32] = acc1;
}

// ---------------------------------------------------------------------------
// In-place row LayerNorm (two-pass for accuracy). One block per row of 512.
// ---------------------------------------------------------------------------
__global__ void layernorm_kernel(float* __restrict__ out,
                                 const float* __restrict__ g,
                                 const float* __restrict__ bta)
{
    __shared__ float red[256];
    const int row = blockIdx.x;
    const int t   = threadIdx.x;
    float* rp = out + (size_t)row * DM_;

    float v0 = rp[t];
    float v1 = rp[t + 256];

    red[t] = v0 + v1;
    __syncthreads();
    for (int st = 128; st > 0; st >>= 1) {
        if (t < st) red[t] += red[t + st];
        __syncthreads();
    }
    const float mu = red[0] * (1.0f / DM_);
    __syncthreads();

    float c0 = v0 - mu, c1 = v1 - mu;
    red[t] = c0 * c0 + c1 * c1;
    __syncthreads();
    for (int st = 128; st > 0; st >>= 1) {
        if (t < st) red[t] += red[t + st];
        __syncthreads();
    }
    const float rstd = rsqrtf(red[0] * (1.0f / DM_) + 1e-6f);
    __syncthreads();

    rp[t]       = c0 * rstd * g[t]       + bta[t];
    rp[t + 256] = c1 * rstd * g[t + 256] + bta[t + 256];
}

// ---------------------------------------------------------------------------
extern "C" void kernel_launch(void* const* d_in, const int* in_sizes, int n_in,
                              void* d_out, int out_size, void* d_ws, size_t ws_size,
                              hipStream_t stream)
{
    const float* hidden = (const float*)d_in[0];
    const float* rpe    = (const float*)d_in[1];
    const int*   qkm    = (const int*)  d_in[2];
    // d_in[3] = k_q_mask (unused by the reference)
    const float* w_q    = (const float*)d_in[4];
    const float* w_k    = (const float*)d_in[5];
    const float* w_v    = (const float*)d_in[6];
    const float* fc_w   = (const float*)d_in[7];
    const float* fc_b   = (const float*)d_in[8];
    const float* ln_g   = (const float*)d_in[9];
    const float* ln_b   = (const float*)d_in[10];
    float* out = (float*)d_out;

    const int M = B_ * S_;        // 4096
    const int N = H_ * D_;        // 512
    const int K = DM_;            // 512
    const size_t per = (size_t)M * N;   // 2,097,152 floats = 8 MiB

    float* ws  = (float*)d_ws;
    float* Qw  = ws;
    float* Kw  = ws + per;
    float* Vw  = ws + 2 * per;
    float* Cw  = ws + 3 * per;

    dim3 ggrid(N / 64, M / 32);   // (8, 128)
    dim3 gblk(32);

    gemm_wmma_f32<<<ggrid, gblk, 0, stream>>>(hidden, w_q, nullptr, nullptr, Qw, M, N, K);
    gemm_wmma_f32<<<ggrid, gblk, 0, stream>>>(hidden, w_k, nullptr, nullptr, Kw, M, N, K);
    gemm_wmma_f32<<<ggrid, gblk, 0, stream>>>(hidden, w_v, nullptr, nullptr, Vw, M, N, K);

    attn_gather_kernel<<<M, 256, 0, stream>>>(Qw, Kw, Vw, rpe, qkm, Cw);

    // out = ctx @ fc_w + fc_b + hidden (fused), then in-place LayerNorm
    gemm_wmma_f32<<<ggrid, gblk, 0, stream>>>(Cw, fc_w, fc_b, hidden, out, M, DM_, N);

    layernorm_kernel<<<M, 256, 0, stream>>>(out, ln_g, ln_b);
}